// TransformerModels_82265803588037
// MI455X (gfx1250) — compile-verified
//
#include <hip/hip_runtime.h>
#include <math.h>

// ---------------------------------------------------------------------------
// Problem constants (match reference)
// ---------------------------------------------------------------------------
#define CV   128      // vocab
#define CD   512      // model dim
#define CL   4        // layers
#define CH   8        // heads
#define CDK  64       // head dim
#define CFF  2048     // ffn dim
#define CM   4        // models
#define CB   2        // batch
#define CT   1024     // seq len
#define CROWS (CM * CB * CT)       // 8192 rows, [m][b][t]
#define CF3  (3 * CD)              // 1536
#define LDSTR 20                   // padded LDS row stride (words)

typedef float v2f __attribute__((ext_vector_type(2)));
typedef float v8f __attribute__((ext_vector_type(8)));

// fp32-exact WMMA: D(16x16,f32) = A(16x4,f32) x B(4x16,f32) + C
static __device__ __forceinline__ v8f wmma_f32(v2f a, v2f b, v8f c) {
  return __builtin_amdgcn_wmma_f32_16x16x4_f32(
      /*neg_a=*/false, a, /*neg_b=*/false, b,
      /*c_mod=*/(short)0, c, /*reuse_a=*/false, /*reuse_b=*/false);
}

// ---------------------------------------------------------------------------
// Embedding: h[m,b,t,d] = tok_emb[m, x[b,t], d] + pos_emb[m, t, d]
// ---------------------------------------------------------------------------
__global__ __launch_bounds__(256) void embed_kernel(
    const int* __restrict__ x, const float* __restrict__ tok,
    const float* __restrict__ pos, float* __restrict__ h) {
  int idx = blockIdx.x * 256 + threadIdx.x;       // 0 .. CROWS*CD
  int d  = idx & (CD - 1);
  int r  = idx >> 9;                              // row = m*2048 + b*1024 + t
  int m  = r >> 11;
  int bt = r & (CB * CT - 1);
  int t  = bt & (CT - 1);
  int tk = x[bt];
  h[idx] = tok[((size_t)m * CV + tk) * CD + d] + pos[((size_t)m * CT + t) * CD + d];
}

// ---------------------------------------------------------------------------
// LayerNorm over D=512: one wave per row (16 elements / lane via float4)
// ---------------------------------------------------------------------------
__global__ __launch_bounds__(256) void ln_kernel(
    const float* __restrict__ X, float* __restrict__ Y,
    const float* __restrict__ W, const float* __restrict__ Bb) {
  const int lane = threadIdx.x & 31;
  const int wave = threadIdx.x >> 5;
  const int row  = blockIdx.x * 8 + wave;
  const int m    = row >> 11;
  const float* x = X + (size_t)row * CD;

  float4 v[4];
#pragma unroll
  for (int i = 0; i < 4; ++i)
    v[i] = *(const float4*)(x + i * 128 + lane * 4);

  float s = 0.f;
#pragma unroll
  for (int i = 0; i < 4; ++i) s += v[i].x + v[i].y + v[i].z + v[i].w;
#pragma unroll
  for (int off = 16; off; off >>= 1) s += __shfl_xor(s, off, 32);
  float mu = s * (1.f / CD);

  float q = 0.f;
#pragma unroll
  for (int i = 0; i < 4; ++i) {
    float a = v[i].x - mu, b = v[i].y - mu, c = v[i].z - mu, d = v[i].w - mu;
    q += a * a + b * b + c * c + d * d;
  }
#pragma unroll
  for (int off = 16; off; off >>= 1) q += __shfl_xor(q, off, 32);
  float rstd = rsqrtf(q * (1.f / CD) + 1e-5f);

  const float* wm = W  + (size_t)m * CD;
  const float* bm = Bb + (size_t)m * CD;
  float* y = Y + (size_t)row * CD;
#pragma unroll
  for (int i = 0; i < 4; ++i) {
    int d0 = i * 128 + lane * 4;
    float4 wv = *(const float4*)(wm + d0);
    float4 bv = *(const float4*)(bm + d0);
    float4 r;
    r.x = (v[i].x - mu) * rstd * wv.x + bv.x;
    r.y = (v[i].y - mu) * rstd * wv.y + bv.y;
    r.z = (v[i].z - mu) * rstd * wv.z + bv.z;
    r.w = (v[i].w - mu) * rstd * wv.w + bv.w;
    *(float4*)(y + d0) = r;
  }
}

// ---------------------------------------------------------------------------
// WMMA GEMM: Y[row, n] = sum_k X[row,k] * W[m][n][k] + bias[m][n]
//   Block: 8 waves x (32 rows) = 256 rows x 64 cols.
//   Weight tile (shared by all 8 waves) staged in LDS, double-buffered,
//   chunk = 16 K-steps. A fragments register-double-buffered per chunk.
//   relu: apply max(0, .)   R: optional residual added   outMode 1: head layout
// ---------------------------------------------------------------------------
__global__ __launch_bounds__(256) void gemm_kernel(
    const float* __restrict__ X, const float* __restrict__ Wt,
    const float* __restrict__ bias, const float* __restrict__ R,
    float* __restrict__ Y, int N, int K, int relu, int outMode) {
  __shared__ float ldsW[2][64 * LDSTR];   // 2 x 5 KB

  const int tid  = threadIdx.x;
  const int lane = tid & 31;
  const int wave = tid >> 5;
  const int half = lane >> 4;
  const int lm   = lane & 15;
  const int row0 = (blockIdx.x * 8 + wave) * 32;   // 2048 % 256 == 0
  const int m    = row0 >> 11;
  const int c0   = blockIdx.y * 64;

  const float* Wm = Wt + (size_t)m * N * K;
  const float* Bm = bias + (size_t)m * N;

  // per-wave A pointers (2 row tiles)
  const float* xr0 = X + (size_t)(row0 +  0 + lm) * K + 2 * half;
  const float* xr1 = X + (size_t)(row0 + 16 + lm) * K + 2 * half;

  // cooperative W staging: thread -> (col, 4-k quad)
  const int scol = tid >> 2;               // 0..63
  const int skq  = (tid & 3) << 2;         // 0,4,8,12
  const float* wsrc = Wm + (size_t)(c0 + scol) * K + skq;
  const int ldst = scol * LDSTR + skq;     // float4-aligned (80B rows)

  v8f acc00 = {0,0,0,0,0,0,0,0};
  v8f acc01 = acc00, acc02 = acc00, acc03 = acc00;
  v8f acc10 = acc00, acc11 = acc00, acc12 = acc00, acc13 = acc00;

  // ---- pipeline prologue: chunk 0 ----
  float4 wreg = *(const float4*)(wsrc);
  v2f Ac[2][4], An[2][4];
#pragma unroll
  for (int ks = 0; ks < 4; ++ks) {
    Ac[0][ks] = *(const v2f*)(xr0 + ks * 4);
    Ac[1][ks] = *(const v2f*)(xr1 + ks * 4);
  }
  *(float4*)(&ldsW[0][ldst]) = wreg;
  __syncthreads();

  const int nk = K >> 4;                   // chunks of 16
  for (int kc = 0; kc < nk; ++kc) {
    const int cur = kc & 1;
    const int kb  = (kc + 1) << 4;
    const bool more = (kc + 1 < nk);
    if (more) {
      wreg = *(const float4*)(wsrc + kb);
#pragma unroll
      for (int ks = 0; ks < 4; ++ks) {
        An[0][ks] = *(const v2f*)(xr0 + kb + ks * 4);
        An[1][ks] = *(const v2f*)(xr1 + kb + ks * 4);
      }
    }

    const float* Lb = &ldsW[cur][0];
#pragma unroll
    for (int ks = 0; ks < 4; ++ks) {
      const int ko = ks * 4 + 2 * half;
      v2f b0 = *(const v2f*)(Lb + (0 * 16 + lm) * LDSTR + ko);
      v2f b1 = *(const v2f*)(Lb + (1 * 16 + lm) * LDSTR + ko);
      v2f b2 = *(const v2f*)(Lb + (2 * 16 + lm) * LDSTR + ko);
      v2f b3 = *(const v2f*)(Lb + (3 * 16 + lm) * LDSTR + ko);
      acc00 = wmma_f32(Ac[0][ks], b0, acc00);
      acc10 = wmma_f32(Ac[1][ks], b0, acc10);
      acc01 = wmma_f32(Ac[0][ks], b1, acc01);
      acc11 = wmma_f32(Ac[1][ks], b1, acc11);
      acc02 = wmma_f32(Ac[0][ks], b2, acc02);
      acc12 = wmma_f32(Ac[1][ks], b2, acc12);
      acc03 = wmma_f32(Ac[0][ks], b3, acc03);
      acc13 = wmma_f32(Ac[1][ks], b3, acc13);
    }

    if (more) {
      *(float4*)(&ldsW[cur ^ 1][ldst]) = wreg;
#pragma unroll
      for (int ks = 0; ks < 4; ++ks) { Ac[0][ks] = An[0][ks]; Ac[1][ks] = An[1][ks]; }
    }
    __syncthreads();
  }

  // ---- epilogue ----
  v8f acc[2][4] = {{acc00, acc01, acc02, acc03}, {acc10, acc11, acc12, acc13}};
#pragma unroll
  for (int rt = 0; rt < 2; ++rt) {
#pragma unroll
    for (int n = 0; n < 4; ++n) {
      const int col = c0 + n * 16 + lm;
      const float bcol = Bm[col];
#pragma unroll
      for (int v = 0; v < 8; ++v) {
        const int row = row0 + rt * 16 + v + 8 * half;
        float val = acc[rt][n][v] + bcol;
        if (relu) val = fmaxf(val, 0.f);
        if (outMode == 0) {
          size_t idx = (size_t)row * N + col;
          if (R) val += R[idx];
          Y[idx] = val;
        } else {  // head: logits[b][m][t][v]
          int b = (row >> 10) & 1;
          int t = row & (CT - 1);
          Y[(((size_t)b * CM + m) * CT + t) * CV + col] = val;
        }
      }
    }
  }
}

// ---------------------------------------------------------------------------
// Causal flash attention, fp32 WMMA. One wave per (m,b,h, 16-query tile).
//   qkv: [m][b][t][h*192 + {q:0..63, k:64..127, v:128..191}]
//   Out: [m][b][t][h*64 + dk]   (head-merged layout, ready for out-proj GEMM)
// ---------------------------------------------------------------------------
__global__ __launch_bounds__(256) void attn_kernel(
    const float* __restrict__ qkv, float* __restrict__ Out) {
  __shared__ float ldsP[8][16 * 16];

  const int lane = threadIdx.x & 31;
  const int wave = threadIdx.x >> 5;
  const int half = lane >> 4;
  const int lm   = lane & 15;
  const int wid   = blockIdx.x * 8 + wave;  // 0..4095
  const int qtile = wid & 63;
  const int combo = wid >> 6;               // 0..63
  const int hh    = combo & 7;
  const int mb    = combo >> 3;             // m*B + b

  const float* base = qkv + (size_t)mb * CT * CF3;
  const int qoff = hh * (3 * CDK);
  const int koff = qoff + CDK;
  const int voff = qoff + 2 * CDK;
  const int q0   = qtile * 16;

  // Preload Q fragments (A-layout): 16 k-steps of 4 over DK=64
  v2f qf[16];
  const float* qr = base + (size_t)(q0 + lm) * CF3 + qoff + 2 * half;
#pragma unroll
  for (int ks = 0; ks < 16; ++ks) qf[ks] = *(const v2f*)(qr + ks * 4);

  v8f o0 = {0,0,0,0,0,0,0,0}, o1 = o0, o2 = o0, o3 = o0;
  float mrow[8], lrow[8];
#pragma unroll
  for (int v = 0; v < 8; ++v) { mrow[v] = -1e30f; lrow[v] = 0.f; }

  float* myP = ldsP[wave];
  const float scale = 0.125f;          // 1/sqrt(64)
  const float log2e = 1.44269504089f;

  for (int j = 0; j <= qtile; ++j) {
    const int t0 = j * 16;

    // ---- load all K fragments first, then serial S accumulation chain ----
    v2f kf[16];
    const float* kr = base + (size_t)(t0 + lm) * CF3 + koff + 2 * half;
#pragma unroll
    for (int ks = 0; ks < 16; ++ks) kf[ks] = *(const v2f*)(kr + ks * 4);

    v8f s = {0,0,0,0,0,0,0,0};
#pragma unroll
    for (int ks = 0; ks < 16; ++ks) s = wmma_f32(qf[ks], kf[ks], s);

    // ---- scale, causal mask, online softmax ----
    const int kcol = t0 + lm;
    float p[8], corr[8];
#pragma unroll
    for (int v = 0; v < 8; ++v) {
      const int qrow = q0 + v + 8 * half;
      float sv = s[v] * scale;
      if (kcol > qrow) sv = -1e30f;
      float red = sv;
#pragma unroll
      for (int off = 8; off; off >>= 1) red = fmaxf(red, __shfl_xor(red, off, 32));
      float mnew = fmaxf(mrow[v], red);
      float pv = exp2f((sv - mnew) * log2e);
      float cv = exp2f((mrow[v] - mnew) * log2e);
      float su = pv;
#pragma unroll
      for (int off = 8; off; off >>= 1) su += __shfl_xor(su, off, 32);
      lrow[v] = lrow[v] * cv + su;
      mrow[v] = mnew;
      corr[v] = cv;
      p[v] = pv;
    }

    // rescale O accumulators
#pragma unroll
    for (int v = 0; v < 8; ++v) {
      o0[v] *= corr[v]; o1[v] *= corr[v]; o2[v] *= corr[v]; o3[v] *= corr[v];
    }

    // stage P (C-layout regs -> row-major LDS) for A-layout re-read
#pragma unroll
    for (int v = 0; v < 8; ++v) myP[(v + 8 * half) * 16 + lm] = p[v];
    asm volatile("" ::: "memory");  // order LDS store->load within wave

    // ---- load all V fragments + P fragments, then O += P @ V ----
    const float* vbase = base + (size_t)t0 * CF3 + voff;
    v2f pa[4], vf[4][4];
#pragma unroll
    for (int ks = 0; ks < 4; ++ks) {
      const int kk = ks * 4 + 2 * half;
      const float* vp = vbase + (size_t)kk * CF3 + lm;
      pa[ks] = *(const v2f*)(&myP[lm * 16 + ks * 4 + 2 * half]);
#pragma unroll
      for (int n = 0; n < 4; ++n) {
        vf[ks][n].x = vp[n * 16];
        vf[ks][n].y = vp[n * 16 + CF3];
      }
    }
#pragma unroll
    for (int ks = 0; ks < 4; ++ks) {
      o0 = wmma_f32(pa[ks], vf[ks][0], o0);
      o1 = wmma_f32(pa[ks], vf[ks][1], o1);
      o2 = wmma_f32(pa[ks], vf[ks][2], o2);
      o3 = wmma_f32(pa[ks], vf[ks][3], o3);
    }
  }

  // ---- normalize and store (head-merged layout) ----
  v8f oo[4] = {o0, o1, o2, o3};
#pragma unroll
  for (int n = 0; n < 4; ++n) {
#pragma unroll
    for (int v = 0; v < 8; ++v) {
      const int t = q0 + v + 8 * half;
      const int d = hh * CDK + n * 16 + lm;
      Out[((size_t)mb * CT + t) * CD + d] = oo[n][v] * (1.f / lrow[v]);
    }
  }
}

// ---------------------------------------------------------------------------
// Host-side orchestration (graph-capture safe: launches only)
// ---------------------------------------------------------------------------
extern "C" void kernel_launch(void* const* d_in, const int* in_sizes, int n_in,
                              void* d_out, int out_size, void* d_ws, size_t ws_size,
                              hipStream_t stream) {
  (void)in_sizes; (void)n_in; (void)out_size; (void)ws_size;
  const int*   x    = (const int*)d_in[0];
  const float* tok  = (const float*)d_in[1];
  const float* pos  = (const float*)d_in[2];
  const float* ln1w = (const float*)d_in[3];
  const float* ln1b = (const float*)d_in[4];
  const float* qkvw = (const float*)d_in[5];
  const float* qkvb = (const float*)d_in[6];
  const float* outw = (const float*)d_in[7];
  const float* outb = (const float*)d_in[8];
  const float* ln2w = (const float*)d_in[9];
  const float* ln2b = (const float*)d_in[10];
  const float* ff1w = (const float*)d_in[11];
  const float* ff1b = (const float*)d_in[12];
  const float* ff2w = (const float*)d_in[13];
  const float* ff2b = (const float*)d_in[14];
  const float* lnfw = (const float*)d_in[15];
  const float* lnfb = (const float*)d_in[16];
  const float* hdw  = (const float*)d_in[17];
  const float* hdb  = (const float*)d_in[18];

  // workspace layout (floats): h[4M] n[4M] qkv[12M] o[4M]; f[16M] aliases qkv+o
  float* ws  = (float*)d_ws;
  float* h   = ws;
  float* n   = ws + (size_t)4  * 1024 * 1024;
  float* qkv = ws + (size_t)8  * 1024 * 1024;
  float* o   = ws + (size_t)20 * 1024 * 1024;
  float* f   = ws + (size_t)8  * 1024 * 1024;   // qkv+o dead when ffn runs

  embed_kernel<<<CROWS * CD / 256, 256, 0, stream>>>(x, tok, pos, h);

  for (int l = 0; l < CL; ++l) {
    ln_kernel<<<CROWS / 8, 256, 0, stream>>>(h, n, ln1w + (size_t)l * CM * CD,
                                             ln1b + (size_t)l * CM * CD);
    gemm_kernel<<<dim3(CROWS / 256, CF3 / 64), 256, 0, stream>>>(
        n, qkvw + (size_t)l * CM * CF3 * CD, qkvb + (size_t)l * CM * CF3,
        nullptr, qkv, CF3, CD, 0, 0);
    attn_kernel<<<(CM * CB * CH * (CT / 16)) / 8, 256, 0, stream>>>(qkv, o);
    gemm_kernel<<<dim3(CROWS / 256, CD / 64), 256, 0, stream>>>(
        o, outw + (size_t)l * CM * CD * CD, outb + (size_t)l * CM * CD,
        h, h, CD, CD, 0, 0);
    ln_kernel<<<CROWS / 8, 256, 0, stream>>>(h, n, ln2w + (size_t)l * CM * CD,
                                             ln2b + (size_t)l * CM * CD);
    gemm_kernel<<<dim3(CROWS / 256, CFF / 64), 256, 0, stream>>>(
        n, ff1w + (size_t)l * CM * CFF * CD, ff1b + (size_t)l * CM * CFF,
        nullptr, f, CFF, CD, 1, 0);
    gemm_kernel<<<dim3(CROWS / 256, CD / 64), 256, 0, stream>>>(
        f, ff2w + (size_t)l * CM * CD * CFF, ff2b + (size_t)l * CM * CD,
        h, h, CD, CFF, 0, 0);
  }

  ln_kernel<<<CROWS / 8, 256, 0, stream>>>(h, n, lnfw, lnfb);
  gemm_kernel<<<dim3(CROWS / 256, CV / 64), 256, 0, stream>>>(
      n, hdw, hdb, nullptr, (float*)d_out, CV, CD, 0, 1);
}